// GAT_70136815944017
// MI455X (gfx1250) — compile-verified
//
#include <hip/hip_runtime.h>
#include <hip/hip_bf16.h>
#include <math.h>

#define IN_CH    128
#define OUT_CH   64
#define N_CLASS  10
#define N_GRAPHS 64
#define NEG_SLOPE 0.2f

typedef float v2f __attribute__((ext_vector_type(2)));
typedef float v8f __attribute__((ext_vector_type(8)));

// ---------------------------------------------------------------------------
// float atomic max via int punning (monotonic int order trick). Works for
// mixed-sign values; used with -inf init (segment max) and 0 init (pool max).
// ---------------------------------------------------------------------------
__device__ __forceinline__ void atomicMaxF32(float* addr, float val) {
    if (val >= 0.0f) {
        atomicMax((int*)addr, __float_as_int(val));
    } else {
        atomicMin((unsigned int*)addr, __float_as_uint(val));
    }
}

__device__ __forceinline__ float leaky(float v) {
    return v > 0.0f ? v : NEG_SLOPE * v;
}

// ---------------------------------------------------------------------------
// Kernel 1: h = x @ W  via V_WMMA_F32_16X16X4_F32 (exact fp32).
// W (32 KB) is staged block-wide into LDS with gfx1250 async global->LDS
// copies (ASYNCcnt-tracked), then B operands come from ds_load while the
// vector-memory path streams x rows. One wave -> 16x64 output tile.
// A 16x4:  lanes 0-15 rows, a.x=K0 a.y=K1 ; lanes 16-31 rows, a.x=K2 a.y=K3
// B 4x16:  lane%16 = column,  b.x/b.y = same K pairs as A per lane-half
// C/D:     VGPR j: lanes 0-15 -> (M=j, N=lane), lanes 16-31 -> (M=j+8, N=lane-16)
// ---------------------------------------------------------------------------
__global__ void gat_gemm_h(const float* __restrict__ x,
                           const float* __restrict__ W,
                           float* __restrict__ h, int N) {
    __shared__ float Wl[IN_CH * OUT_CH];          // 32 KB

    // ---- async stage W into LDS (2048 b128 transfers, 8 per thread) ----
    {
        const unsigned int lds0 = (unsigned int)(uintptr_t)&Wl[0]; // flat low 32 = LDS addr
        for (int t = threadIdx.x; t < (IN_CH * OUT_CH) / 4; t += blockDim.x) {
            unsigned int       ldsaddr = lds0 + (unsigned int)t * 16u;
            unsigned long long gaddr   = (unsigned long long)(const void*)(W + (size_t)t * 4);
            asm volatile("global_load_async_to_lds_b128 %0, %1, off"
                         :: "v"(ldsaddr), "v"(gaddr) : "memory");
        }
        asm volatile("s_wait_asynccnt 0x0" ::: "memory");
    }
    __syncthreads();

    const int wave = threadIdx.x >> 5;
    const int lane = threadIdx.x & 31;
    const int row0 = (blockIdx.x * 8 + wave) * 16;
    if (row0 >= N) return;                      // wave-uniform: EXEC stays all-1s

    const int lrow  = lane & 15;
    const int khalf = lane >> 4;                // 0 or 1
    int r = row0 + lrow;
    if (r >= N) r = N - 1;                      // clamp; store is guarded below
    const float* xr = x + (size_t)r * IN_CH;

    v8f acc0 = {}, acc1 = {}, acc2 = {}, acc3 = {};
    for (int k = 0; k < IN_CH; k += 4) {
        if (k + 32 < IN_CH) __builtin_prefetch(xr + k + 32, 0, 3);
        v2f a;
        a.x = xr[k + 2 * khalf + 0];
        a.y = xr[k + 2 * khalf + 1];
        const float* w0 = &Wl[(k + 2 * khalf) * OUT_CH];          // row K   (LDS)
        const float* w1 = w0 + OUT_CH;                            // row K+1 (LDS)
        v2f b0, b1, b2, b3;
        b0.x = w0[lrow +  0]; b0.y = w1[lrow +  0];
        b1.x = w0[lrow + 16]; b1.y = w1[lrow + 16];
        b2.x = w0[lrow + 32]; b2.y = w1[lrow + 32];
        b3.x = w0[lrow + 48]; b3.y = w1[lrow + 48];
        acc0 = __builtin_amdgcn_wmma_f32_16x16x4_f32(false, a, false, b0, (short)0, acc0, false, false);
        acc1 = __builtin_amdgcn_wmma_f32_16x16x4_f32(false, a, false, b1, (short)0, acc1, false, false);
        acc2 = __builtin_amdgcn_wmma_f32_16x16x4_f32(false, a, false, b2, (short)0, acc2, false, false);
        acc3 = __builtin_amdgcn_wmma_f32_16x16x4_f32(false, a, false, b3, (short)0, acc3, false, false);
    }

    const int srow_off = khalf ? 8 : 0;
#pragma unroll
    for (int j = 0; j < 8; ++j) {
        const int rr = row0 + j + srow_off;
        if (rr < N) {
            float* hr = h + (size_t)rr * OUT_CH;
            hr[ 0 + lrow] = acc0[j];
            hr[16 + lrow] = acc1[j];
            hr[32 + lrow] = acc2[j];
            hr[48 + lrow] = acc3[j];
        }
    }
}

// ---------------------------------------------------------------------------
// Kernel 2: per-node attention logits a_s = h.att_src, a_d = h.att_dst
// ---------------------------------------------------------------------------
__global__ void gat_node_att(const float* __restrict__ h,
                             const float* __restrict__ att_src,
                             const float* __restrict__ att_dst,
                             float* __restrict__ a_s, float* __restrict__ a_d,
                             int N) {
    int i = blockIdx.x * blockDim.x + threadIdx.x;
    if (i >= N) return;
    const float* hi = h + (size_t)i * OUT_CH;
    float s = 0.0f, d = 0.0f;
#pragma unroll 8
    for (int c = 0; c < OUT_CH; ++c) {
        float v = hi[c];
        s += v * att_src[c];
        d += v * att_dst[c];
    }
    a_s[i] = s;
    a_d[i] = d;
}

// ---------------------------------------------------------------------------
// Kernel 3: workspace init (out_raw=0, m=-inf, denom=0, pools=0, cnt=0)
// ---------------------------------------------------------------------------
__global__ void gat_init(float* __restrict__ out_raw, float* __restrict__ m,
                         float* __restrict__ denom, float* __restrict__ pool_max,
                         float* __restrict__ pool_sum, float* __restrict__ cnt,
                         int N) {
    int i = blockIdx.x * blockDim.x + threadIdx.x;
    const int total = N * OUT_CH;
    if (i < total) out_raw[i] = 0.0f;
    if (i < N) { m[i] = -INFINITY; denom[i] = 0.0f; }
    if (i < N_GRAPHS * OUT_CH) { pool_max[i] = 0.0f; pool_sum[i] = 0.0f; }
    if (i < N_GRAPHS) cnt[i] = 0.0f;
}

__device__ __forceinline__ void edge_endpoints(const int* __restrict__ ei,
                                               int e, int E, int& src, int& dst) {
    if (e < E) { src = ei[e]; dst = ei[E + e]; }
    else       { src = e - E; dst = e - E; }          // self loop
}

// ---------------------------------------------------------------------------
// Kernel 4: segment max of leaky_relu(a_s[src] + a_d[dst]) over dst
// ---------------------------------------------------------------------------
__global__ void gat_edge_max(const int* __restrict__ ei,
                             const float* __restrict__ a_s,
                             const float* __restrict__ a_d,
                             float* __restrict__ m, int E, int Etot) {
    int e = blockIdx.x * blockDim.x + threadIdx.x;
    if (e >= Etot) return;
    int src, dst; edge_endpoints(ei, e, E, src, dst);
    atomicMaxF32(&m[dst], leaky(a_s[src] + a_d[dst]));
}

// ---------------------------------------------------------------------------
// Kernel 5: denom[dst] += exp(e - m[dst])
// ---------------------------------------------------------------------------
__global__ void gat_edge_expsum(const int* __restrict__ ei,
                                const float* __restrict__ a_s,
                                const float* __restrict__ a_d,
                                const float* __restrict__ m,
                                float* __restrict__ denom, int E, int Etot) {
    int e = blockIdx.x * blockDim.x + threadIdx.x;
    if (e >= Etot) return;
    int src, dst; edge_endpoints(ei, e, E, src, dst);
    float mv = m[dst];
    if (!isfinite(mv)) mv = 0.0f;
    atomicAdd(&denom[dst], expf(leaky(a_s[src] + a_d[dst]) - mv));
}

// ---------------------------------------------------------------------------
// Kernel 6: out_raw[dst] += alpha * h[src]   (one wave per edge, 2 ch / lane)
// ---------------------------------------------------------------------------
__global__ void gat_edge_aggregate(const int* __restrict__ ei,
                                   const float* __restrict__ a_s,
                                   const float* __restrict__ a_d,
                                   const float* __restrict__ m,
                                   const float* __restrict__ denom,
                                   const float* __restrict__ h,
                                   float* __restrict__ out_raw,
                                   int E, int Etot) {
    const int wave = threadIdx.x >> 5;
    const int lane = threadIdx.x & 31;
    const int e = blockIdx.x * 8 + wave;
    if (e >= Etot) return;
    int src, dst; edge_endpoints(ei, e, E, src, dst);
    float mv = m[dst];
    if (!isfinite(mv)) mv = 0.0f;
    const float alpha = expf(leaky(a_s[src] + a_d[dst]) - mv) / (denom[dst] + 1e-16f);
    const float* hs = h + (size_t)src * OUT_CH;
    float* od = out_raw + (size_t)dst * OUT_CH;
    const int c = 2 * lane;
    atomicAdd(&od[c + 0], alpha * hs[c + 0]);
    atomicAdd(&od[c + 1], alpha * hs[c + 1]);
}

// ---------------------------------------------------------------------------
// Kernel 7: +bias, L2-normalize, ReLU, then pooled max/sum/count per graph.
// One wave per node; 32-lane shfl_xor butterfly for the norm.
// ---------------------------------------------------------------------------
__global__ void gat_norm_pool(const float* __restrict__ out_raw,
                              const float* __restrict__ bias,
                              const int* __restrict__ batch,
                              float* __restrict__ pool_max,
                              float* __restrict__ pool_sum,
                              float* __restrict__ cnt, int N) {
    const int wave = threadIdx.x >> 5;
    const int lane = threadIdx.x & 31;
    const int i = blockIdx.x * 8 + wave;
    if (i >= N) return;
    const int c = 2 * lane;
    float v0 = out_raw[(size_t)i * OUT_CH + c + 0] + bias[c + 0];
    float v1 = out_raw[(size_t)i * OUT_CH + c + 1] + bias[c + 1];
    float ss = v0 * v0 + v1 * v1;
#pragma unroll
    for (int o = 16; o > 0; o >>= 1) ss += __shfl_xor(ss, o, 32);
    const float inv = 1.0f / fmaxf(sqrtf(ss), 1e-12f);
    v0 = fmaxf(v0 * inv, 0.0f);
    v1 = fmaxf(v1 * inv, 0.0f);
    const int g = batch[i];
    atomicMaxF32(&pool_max[g * OUT_CH + c + 0], v0);
    atomicMaxF32(&pool_max[g * OUT_CH + c + 1], v1);
    atomicAdd(&pool_sum[g * OUT_CH + c + 0], v0);
    atomicAdd(&pool_sum[g * OUT_CH + c + 1], v1);
    if (lane == 0) atomicAdd(&cnt[g], 1.0f);
}

// ---------------------------------------------------------------------------
// Kernel 8: logits = [max | mean] @ lin_w + lin_b   (64x128x10, trivial)
// ---------------------------------------------------------------------------
__global__ void gat_final(const float* __restrict__ pool_max,
                          const float* __restrict__ pool_sum,
                          const float* __restrict__ cnt,
                          const float* __restrict__ lin_w,
                          const float* __restrict__ lin_b,
                          float* __restrict__ out) {
    int idx = blockIdx.x * blockDim.x + threadIdx.x;
    if (idx >= N_GRAPHS * N_CLASS) return;
    const int g = idx / N_CLASS;
    const int c = idx % N_CLASS;
    const float inv_cnt = 1.0f / fmaxf(cnt[g], 1.0f);
    float acc = lin_b[c];
#pragma unroll 8
    for (int k = 0; k < OUT_CH; ++k) {
        acc += pool_max[g * OUT_CH + k] * lin_w[k * N_CLASS + c];
        acc += (pool_sum[g * OUT_CH + k] * inv_cnt) * lin_w[(OUT_CH + k) * N_CLASS + c];
    }
    out[g * N_CLASS + c] = acc;
}

// ---------------------------------------------------------------------------
extern "C" void kernel_launch(void* const* d_in, const int* in_sizes, int n_in,
                              void* d_out, int out_size, void* d_ws, size_t ws_size,
                              hipStream_t stream) {
    const float* x       = (const float*)d_in[0];
    const int*   ei      = (const int*)d_in[1];
    const int*   batch   = (const int*)d_in[2];
    const float* W       = (const float*)d_in[3];
    const float* att_src = (const float*)d_in[4];
    const float* att_dst = (const float*)d_in[5];
    const float* bias    = (const float*)d_in[6];
    const float* lin_w   = (const float*)d_in[7];
    const float* lin_b   = (const float*)d_in[8];
    float* out = (float*)d_out;

    const int N    = in_sizes[0] / IN_CH;
    const int E    = in_sizes[1] / 2;
    const int Etot = E + N;        // + self loops

    // workspace carve-up (floats)
    float* ws       = (float*)d_ws;
    float* h        = ws;                               // N*64
    float* out_raw  = h + (size_t)N * OUT_CH;           // N*64
    float* a_s      = out_raw + (size_t)N * OUT_CH;     // N
    float* a_d      = a_s + N;                          // N
    float* m        = a_d + N;                          // N
    float* denom    = m + N;                            // N
    float* pool_max = denom + N;                        // 64*64
    float* pool_sum = pool_max + N_GRAPHS * OUT_CH;     // 64*64
    float* cnt      = pool_sum + N_GRAPHS * OUT_CH;     // 64

    const int TB = 256;

    gat_gemm_h<<<(N + 127) / 128, TB, 0, stream>>>(x, W, h, N);
    gat_node_att<<<(N + TB - 1) / TB, TB, 0, stream>>>(h, att_src, att_dst, a_s, a_d, N);
    gat_init<<<((size_t)N * OUT_CH + TB - 1) / TB, TB, 0, stream>>>(out_raw, m, denom,
                                                                    pool_max, pool_sum, cnt, N);
    gat_edge_max<<<(Etot + TB - 1) / TB, TB, 0, stream>>>(ei, a_s, a_d, m, E, Etot);
    gat_edge_expsum<<<(Etot + TB - 1) / TB, TB, 0, stream>>>(ei, a_s, a_d, m, denom, E, Etot);
    gat_edge_aggregate<<<(Etot + 7) / 8, TB, 0, stream>>>(ei, a_s, a_d, m, denom, h,
                                                          out_raw, E, Etot);
    gat_norm_pool<<<(N + 7) / 8, TB, 0, stream>>>(out_raw, bias, batch,
                                                  pool_max, pool_sum, cnt, N);
    gat_final<<<(N_GRAPHS * N_CLASS + TB - 1) / TB, TB, 0, stream>>>(pool_max, pool_sum, cnt,
                                                                     lin_w, lin_b, out);
}